// TransE_20409684590819
// MI455X (gfx1250) — compile-verified
//
#include <hip/hip_runtime.h>
#include <math.h>

#define NENT 50000
#define DIM 128
#define BATCH 64
#define TN 32            // entities per tile
#define TSTR 132         // padded LDS row stride (floats): conflict-free float4 reads
#define DIST_BLOCK 256
#define DIST_GRID 512

// ---------------------------------------------------------------------------
// Async tile loader: one tile = TN x DIM f32 of ent_emb -> LDS (padded rows).
// Each wave issues 4 x global_load_async_to_lds_b128 (32 lanes x 16B = 512B each),
// tracked by ASYNCcnt.
// ---------------------------------------------------------------------------
__device__ __forceinline__ void issue_tile(const float* __restrict__ ent,
                                           unsigned ldsbase, int tile, int tid) {
    const int n0 = tile * TN;
#pragma unroll
    for (int j = 0; j < 4; ++j) {
        int slot = j * DIST_BLOCK + tid;      // 0..1023 = 32 rows x 32 float4-cols
        int row  = slot >> 5;
        int c4   = slot & 31;
        int n    = n0 + row;
        n = (n < NENT) ? n : (NENT - 1);      // clamp tail: never read OOB
        unsigned go = (unsigned)((n * DIM + c4 * 4) * 4);              // global byte off
        unsigned lo = ldsbase + (unsigned)((row * TSTR + c4 * 4) * 4); // LDS byte addr
        asm volatile("global_load_async_to_lds_b128 %0, %1, %2"
                     :: "v"(lo), "v"(go), "s"(ent) : "memory");
    }
}

// ---------------------------------------------------------------------------
// Kernel 2 (emitted first for disasm visibility):
// dist[b][n] = sum_d | head[b][d] - normalize(ent[n])[d] |
// Double-buffered async global->LDS pipeline under ASYNCcnt; the head matrix
// staging is folded into the same async pipeline (in-order retirement means
// the first  s_wait_asynccnt 4  drains head + tile0 together).
// ---------------------------------------------------------------------------
__global__ __launch_bounds__(DIST_BLOCK)
void dist_kernel(const float* __restrict__ ent, const float* __restrict__ head,
                 float* __restrict__ out) {
    __shared__ float sh_head[BATCH * DIM];                       // 32 KB
    __shared__ __align__(16) float sh_tile[2][TN * TSTR];        // 2 x 16.5 KB
    __shared__ float sh_scale[TN];
    __shared__ float sh_psum[TN * 8];

    const int tid  = threadIdx.x;
    const int wave = tid >> 5;
    const int lane = tid & 31;

    unsigned headbase = (unsigned)(unsigned long long)&sh_head[0];
    unsigned ldsbase[2];
    ldsbase[0] = (unsigned)(unsigned long long)&sh_tile[0][0];   // low 32b of flat = LDS off
    ldsbase[1] = (unsigned)(unsigned long long)&sh_tile[1][0];

    // stage head matrix into LDS via async copy (8 x b128 per wave = 32 KB total)
#pragma unroll
    for (int j = 0; j < BATCH * DIM / 4 / DIST_BLOCK; ++j) {     // 8 iterations
        unsigned off = (unsigned)((j * DIST_BLOCK + tid) * 16);  // float4 slots
        asm volatile("global_load_async_to_lds_b128 %0, %1, %2"
                     :: "v"(headbase + off), "v"(off), "s"(head) : "memory");
    }

    const int ntiles = (NENT + TN - 1) / TN;
    const int G = gridDim.x;
    const int tile0 = blockIdx.x;
    if (tile0 >= ntiles) return;

    issue_tile(ent, ldsbase[0], tile0, tid);                     // prologue prefetch

    int buf = 0;
    for (int t = tile0; t < ntiles; t += G) {
        const int nt = t + G;
        if (nt < ntiles) {
            issue_tile(ent, ldsbase[buf ^ 1], nt, tid);          // prefetch next tile
            asm volatile("s_wait_asynccnt 4" ::: "memory");      // all but next-tile done
        } else {
            asm volatile("s_wait_asynccnt 0" ::: "memory");      // drain
        }
        __syncthreads();                                         // tile t (+head) visible

        // --- normalize tile rows in place: 8 threads per row ---
        const int row = tid >> 3, seg = tid & 7;
        float* rp = &sh_tile[buf][row * TSTR + seg * 16];
        float ss = 0.f;
#pragma unroll
        for (int e = 0; e < 16; ++e) { float v = rp[e]; ss += v * v; }
        sh_psum[tid] = ss;                                       // tid == row*8+seg
        __syncthreads();
        if (tid < TN) {
            float s = 0.f;
#pragma unroll
            for (int j = 0; j < 8; ++j) s += sh_psum[tid * 8 + j];
            sh_scale[tid] = 1.0f / fmaxf(sqrtf(s), 1e-12f);
        }
        __syncthreads();
        {
            float scl = sh_scale[row];
#pragma unroll
            for (int e = 0; e < 16; ++e) rp[e] *= scl;
        }
        __syncthreads();

        // --- L1 distances: wave w -> batch rows [8w,8w+8), lane -> entity n0+lane ---
        const int n = t * TN + lane;
        float acc[8];
#pragma unroll
        for (int i = 0; i < 8; ++i) acc[i] = 0.f;
#pragma unroll 4
        for (int d4 = 0; d4 < DIM / 4; ++d4) {
            float4 xv = *(const float4*)&sh_tile[buf][lane * TSTR + d4 * 4];
#pragma unroll
            for (int i = 0; i < 8; ++i) {
                float4 hv = *(const float4*)&sh_head[(wave * 8 + i) * DIM + d4 * 4]; // broadcast
                acc[i] += fabsf(hv.x - xv.x) + fabsf(hv.y - xv.y) +
                          fabsf(hv.z - xv.z) + fabsf(hv.w - xv.w);
            }
        }
        if (n < NENT) {
#pragma unroll
            for (int i = 0; i < 8; ++i)
                out[(size_t)(wave * 8 + i) * NENT + n] = acc[i];
        }
        __syncthreads();     // all reads of buf done before it is refilled next iter
        buf ^= 1;
    }
}

// ---------------------------------------------------------------------------
// Kernel 1: head[b][d] = normalize(ent[e1[b]]) + normalize(rel_emb[rel[b]])
// ---------------------------------------------------------------------------
__global__ void head_kernel(const int* __restrict__ e1, const int* __restrict__ rel,
                            const float* __restrict__ ent, const float* __restrict__ rle,
                            float* __restrict__ head) {
    const int b = blockIdx.x;
    const int d = threadIdx.x;              // 128 threads
    float e = ent[(size_t)e1[b] * DIM + d];
    float r = rle[(size_t)rel[b] * DIM + d];
    __shared__ float re[DIM];
    __shared__ float rr[DIM];
    re[d] = e * e;
    rr[d] = r * r;
    __syncthreads();
    for (int s = DIM / 2; s > 0; s >>= 1) {
        if (d < s) { re[d] += re[d + s]; rr[d] += rr[d + s]; }
        __syncthreads();
    }
    float ne = 1.0f / fmaxf(sqrtf(re[0]), 1e-12f);
    float nr = 1.0f / fmaxf(sqrtf(rr[0]), 1e-12f);
    head[b * DIM + d] = e * ne + r * nr;
}

// ---------------------------------------------------------------------------
// Kernel 3: per-row max and sum(exp(x - max))  (64 rows)
// ---------------------------------------------------------------------------
__global__ void rowstat_kernel(const float* __restrict__ dist, float* __restrict__ stats) {
    const int r = blockIdx.x;
    const int tid = threadIdx.x;
    const int BS = 256;
    __shared__ float red[256];
    const float* row = dist + (size_t)r * NENT;

    float m = -INFINITY;
    for (int i = tid; i < NENT; i += BS) m = fmaxf(m, row[i]);
    red[tid] = m;
    __syncthreads();
    for (int s = 128; s > 0; s >>= 1) {
        if (tid < s) red[tid] = fmaxf(red[tid], red[tid + s]);
        __syncthreads();
    }
    const float M = red[0];
    __syncthreads();

    float sum = 0.f;
    for (int i = tid; i < NENT; i += BS) sum += expf(row[i] - M);
    red[tid] = sum;
    __syncthreads();
    for (int s = 128; s > 0; s >>= 1) {
        if (tid < s) red[tid] += red[tid + s];
        __syncthreads();
    }
    if (tid == 0) { stats[2 * r] = M; stats[2 * r + 1] = red[0]; }
}

// ---------------------------------------------------------------------------
// Kernel 4: out = exp(dist - max) / sum, in place, float4 (NENT % 4 == 0)
// ---------------------------------------------------------------------------
__global__ void softmax_kernel(float* __restrict__ out, const float* __restrict__ stats) {
    const int r = blockIdx.y;
    const int g = blockIdx.x * blockDim.x + threadIdx.x;   // float4 index in row
    if (g >= NENT / 4) return;
    const float M   = stats[2 * r];
    const float inv = 1.0f / stats[2 * r + 1];
    float4* p = (float4*)(out + (size_t)r * NENT) + g;
    float4 v = *p;
    v.x = expf(v.x - M) * inv;
    v.y = expf(v.y - M) * inv;
    v.z = expf(v.z - M) * inv;
    v.w = expf(v.w - M) * inv;
    *p = v;
}

// ---------------------------------------------------------------------------
extern "C" void kernel_launch(void* const* d_in, const int* in_sizes, int n_in,
                              void* d_out, int out_size, void* d_ws, size_t ws_size,
                              hipStream_t stream) {
    const int*   e1  = (const int*)d_in[0];
    const int*   rel = (const int*)d_in[1];
    const float* ent = (const float*)d_in[4];
    const float* rle = (const float*)d_in[5];
    float* out = (float*)d_out;

    float* head  = (float*)d_ws;           // 64*128 floats = 32 KB
    float* stats = head + BATCH * DIM;     // 128 floats (max,sum per row)

    head_kernel<<<BATCH, DIM, 0, stream>>>(e1, rel, ent, rle, head);
    dist_kernel<<<DIST_GRID, DIST_BLOCK, 0, stream>>>(ent, head, out);
    rowstat_kernel<<<BATCH, 256, 0, stream>>>(out, stats);
    dim3 g((NENT / 4 + 255) / 256, BATCH);
    softmax_kernel<<<g, 256, 0, stream>>>(out, stats);
}